// TransformerConvModel_51857435132128
// MI455X (gfx1250) — compile-verified
//
#include <hip/hip_runtime.h>
#include <math.h>

#define NNODES 50000
#define NEDGES 800000
#define HIDD   128
#define NHEAD  4
#define NB     64
#define NCLS   2

typedef __attribute__((ext_vector_type(2))) float v2f;
typedef __attribute__((ext_vector_type(8))) float v8f;

__device__ __forceinline__ void atomicMaxF(float* addr, float val) {
    // monotonic float ordering: signed max for >=0, unsigned min for <0.
    if (val >= 0.0f) atomicMax((int*)addr, __float_as_int(val));
    else             atomicMin((unsigned int*)addr, __float_as_uint(val));
}

// ---------------- fused Q/K/V/Skip GEMM: out_m = x @ W_m + b_m -------------
// grid = N/16 row strips, block = 256 (8 waves). Wave w computes columns
// [16w,16w+16) of all four outputs via V_WMMA_F32_16X16X4_F32.
__global__ __launch_bounds__(256)
void gemm4_wmma(const float* __restrict__ x,
                const float* __restrict__ W0, const float* __restrict__ b0,
                const float* __restrict__ W1, const float* __restrict__ b1,
                const float* __restrict__ W2, const float* __restrict__ b2,
                const float* __restrict__ W3, const float* __restrict__ b3,
                float* __restrict__ o0, float* __restrict__ o1,
                float* __restrict__ o2, float* __restrict__ o3) {
    __shared__ float xs[16][132];                 // padded: stride 132 kills bank conflicts
    const int i0 = blockIdx.x * 16;

    for (int t = threadIdx.x; t < 512; t += 256) {         // 16x128 floats as float4
        const int r = t >> 5, c4 = t & 31;
        float4 vv = ((const float4*)(x + (size_t)(i0 + r) * HIDD))[c4];
        xs[r][c4*4+0] = vv.x; xs[r][c4*4+1] = vv.y;
        xs[r][c4*4+2] = vv.z; xs[r][c4*4+3] = vv.w;
    }
    __syncthreads();

    const int wave = threadIdx.x >> 5;
    const int lane = threadIdx.x & 31;
    const int j0   = wave * 16;
    const int rc   = lane & 15;     // A row / B,C,D column
    const int kh   = lane >> 4;     // which K-half this lane holds

    v8f c0 = {}, c1 = {}, c2 = {}, c3 = {};
    #pragma unroll
    for (int k0 = 0; k0 < HIDD; k0 += 4) {
        v2f a;                                   // A 16x4 f32 fragment
        a.x = xs[rc][k0 + 2*kh + 0];
        a.y = xs[rc][k0 + 2*kh + 1];
        const int boff = (k0 + 2*kh) * HIDD + j0 + rc;
        v2f bb;
        bb.x = W0[boff]; bb.y = W0[boff + HIDD];
        c0 = __builtin_amdgcn_wmma_f32_16x16x4_f32(false, a, false, bb, (short)0, c0, false, false);
        bb.x = W1[boff]; bb.y = W1[boff + HIDD];
        c1 = __builtin_amdgcn_wmma_f32_16x16x4_f32(false, a, false, bb, (short)0, c1, false, false);
        bb.x = W2[boff]; bb.y = W2[boff + HIDD];
        c2 = __builtin_amdgcn_wmma_f32_16x16x4_f32(false, a, false, bb, (short)0, c2, false, false);
        bb.x = W3[boff]; bb.y = W3[boff + HIDD];
        c3 = __builtin_amdgcn_wmma_f32_16x16x4_f32(false, a, false, bb, (short)0, c3, false, false);
    }
    const float B0 = b0[j0+rc], B1 = b1[j0+rc], B2 = b2[j0+rc], B3 = b3[j0+rc];
    #pragma unroll
    for (int r = 0; r < 8; ++r) {
        const size_t o = (size_t)(i0 + r + 8*kh) * HIDD + j0 + rc;
        o0[o] = c0[r] + B0;  o1[o] = c1[r] + B1;
        o2[o] = c2[r] + B2;  o3[o] = c3[r] + B3;
    }
}

// ---------------- per-layer scratch init -----------------------------------
__global__ void init_layer(float* agg, float* amax, float* denom) {
    int t = blockIdx.x * blockDim.x + threadIdx.x;
    if (t < NNODES * HIDD) agg[t] = 0.0f;
    if (t < NNODES * NHEAD) { amax[t] = -INFINITY; denom[t] = 0.0f; }
}

// ---------------- edge attention logits + running segment max --------------
__global__ void edge_logits(const float* __restrict__ q, const float* __restrict__ k,
                            const int* __restrict__ src, const int* __restrict__ dst,
                            float* __restrict__ alpha, float* __restrict__ amax) {
    int e = blockIdx.x * blockDim.x + threadIdx.x;
    if (e >= NEDGES) return;
    const int s = src[e], d = dst[e];
    const float4* qd = (const float4*)(q + (size_t)d * HIDD);
    const float4* ks = (const float4*)(k + (size_t)s * HIDD);
    #pragma unroll
    for (int h = 0; h < NHEAD; ++h) {
        float acc = 0.0f;
        #pragma unroll
        for (int c = 0; c < 8; ++c) {
            float4 a = qd[h*8 + c], b = ks[h*8 + c];
            acc += a.x*b.x + a.y*b.y + a.z*b.z + a.w*b.w;
        }
        const float al = acc * 0.17677669529663687f;   // 1/sqrt(32)
        alpha[(size_t)e * NHEAD + h] = al;
        atomicMaxF(&amax[d * NHEAD + h], al);
    }
}

// ---------------- exp(alpha - max) + segment-sum denominator ---------------
__global__ void edge_exp(float* __restrict__ alpha, const int* __restrict__ dst,
                         const float* __restrict__ amax, float* __restrict__ denom) {
    int e = blockIdx.x * blockDim.x + threadIdx.x;
    if (e >= NEDGES) return;
    const int d = dst[e];
    #pragma unroll
    for (int h = 0; h < NHEAD; ++h) {
        const float ex = expf(alpha[(size_t)e*NHEAD + h] - amax[d*NHEAD + h]);
        alpha[(size_t)e*NHEAD + h] = ex;
        atomicAdd(&denom[d*NHEAD + h], ex);
    }
}

// ---------------- attention-weighted scatter of V --------------------------
__global__ void edge_aggregate(const float* __restrict__ alpha, const float* __restrict__ denom,
                               const float* __restrict__ v, const int* __restrict__ src,
                               const int* __restrict__ dst, float* __restrict__ agg) {
    int t = blockIdx.x * blockDim.x + threadIdx.x;     // one thread per (edge, head)
    if (t >= NEDGES * NHEAD) return;
    const int e = t >> 2, h = t & 3;
    const int s = src[e], d = dst[e];
    const float attn = alpha[t] / (denom[d*NHEAD + h] + 1e-16f);
    const float4* vs = (const float4*)(v + (size_t)s * HIDD + h * 32);
    float* ag = agg + (size_t)d * HIDD + h * 32;
    #pragma unroll
    for (int c = 0; c < 8; ++c) {
        float4 vv = vs[c];
        atomicAdd(&ag[c*4+0], attn * vv.x);
        atomicAdd(&ag[c*4+1], attn * vv.y);
        atomicAdd(&ag[c*4+2], attn * vv.z);
        atomicAdd(&ag[c*4+3], attn * vv.w);
    }
}

// ---------------- h = relu(agg + skip) --------------------------------------
__global__ void finalize_layer(const float* __restrict__ agg, const float* __restrict__ skip,
                               float* __restrict__ h) {
    int t = blockIdx.x * blockDim.x + threadIdx.x;
    if (t >= NNODES * HIDD) return;
    h[t] = fmaxf(agg[t] + skip[t], 0.0f);
}

// ---------------- graph pooling ---------------------------------------------
__global__ void pool_init(float* pool) {
    int t = blockIdx.x * blockDim.x + threadIdx.x;
    if (t >= NB * 2 * HIDD) return;
    pool[t] = ((t & 255) < HIDD) ? 0.0f : -INFINITY;   // [sum(128) | max(128)] per graph
}

__global__ void pool_accumulate(const float* __restrict__ h, const int* __restrict__ batch,
                                float* __restrict__ pool) {
    int t = blockIdx.x * blockDim.x + threadIdx.x;
    if (t >= NNODES * HIDD) return;
    const int n = t / HIDD, f = t % HIDD;
    const int b = batch[n];
    const float val = h[t];
    atomicAdd(&pool[b * 256 + f], val);
    atomicMaxF(&pool[b * 256 + HIDD + f], val);
}

__global__ void classifier(const float* __restrict__ pool, const float* __restrict__ Wlin,
                           const float* __restrict__ blin, float* __restrict__ out) {
    int t = threadIdx.x;                 // 128 threads = 64 graphs x 2 classes
    if (t >= NB * NCLS) return;
    const int b = t / NCLS, c = t % NCLS;
    float acc = blin[c];
    for (int f = 0; f < 2 * HIDD; ++f) {
        float g = pool[b * 256 + f];
        if (f >= HIDD && !isfinite(g)) g = 0.0f;       // empty-segment max -> 0
        acc += g * Wlin[f * NCLS + c];
    }
    out[t] = acc;
}

// ---------------------------------------------------------------------------
static void run_layer(const float* xin,
                      const float* Wq, const float* bq, const float* Wk, const float* bk,
                      const float* Wv, const float* bv, const float* Ws, const float* bs,
                      const int* src, const int* dst,
                      float* q, float* k, float* v, float* s,
                      float* agg, float* alpha, float* amax, float* denom,
                      float* h, hipStream_t stream) {
    const int NT = 256;
    init_layer<<<(NNODES*HIDD + NT-1)/NT, NT, 0, stream>>>(agg, amax, denom);
    gemm4_wmma<<<NNODES/16, NT, 0, stream>>>(xin, Wq,bq, Wk,bk, Wv,bv, Ws,bs, q,k,v,s);
    edge_logits<<<(NEDGES + NT-1)/NT, NT, 0, stream>>>(q, k, src, dst, alpha, amax);
    edge_exp<<<(NEDGES + NT-1)/NT, NT, 0, stream>>>(alpha, dst, amax, denom);
    edge_aggregate<<<(NEDGES*NHEAD + NT-1)/NT, NT, 0, stream>>>(alpha, denom, v, src, dst, agg);
    finalize_layer<<<(NNODES*HIDD + NT-1)/NT, NT, 0, stream>>>(agg, s, h);
}

extern "C" void kernel_launch(void* const* d_in, const int* in_sizes, int n_in,
                              void* d_out, int out_size, void* d_ws, size_t ws_size,
                              hipStream_t stream) {
    const float* x     = (const float*)d_in[0];
    const int*   eidx  = (const int*)d_in[1];
    const int*   batch = (const int*)d_in[2];
    const float *Wq1=(const float*)d_in[3],  *bq1=(const float*)d_in[4];
    const float *Wk1=(const float*)d_in[5],  *bk1=(const float*)d_in[6];
    const float *Wv1=(const float*)d_in[7],  *bv1=(const float*)d_in[8];
    const float *Ws1=(const float*)d_in[9],  *bs1=(const float*)d_in[10];
    const float *Wq2=(const float*)d_in[11], *bq2=(const float*)d_in[12];
    const float *Wk2=(const float*)d_in[13], *bk2=(const float*)d_in[14];
    const float *Wv2=(const float*)d_in[15], *bv2=(const float*)d_in[16];
    const float *Ws2=(const float*)d_in[17], *bs2=(const float*)d_in[18];
    const float *Wlin=(const float*)d_in[19], *blin=(const float*)d_in[20];

    const int* src = eidx;             // edge_index[0]
    const int* dst = eidx + NEDGES;    // edge_index[1]

    float* w = (float*)d_ws;
    float* q     = w; w += (size_t)NNODES * HIDD;
    float* k     = w; w += (size_t)NNODES * HIDD;
    float* v     = w; w += (size_t)NNODES * HIDD;
    float* s     = w; w += (size_t)NNODES * HIDD;
    float* agg   = w; w += (size_t)NNODES * HIDD;
    float* h     = w; w += (size_t)NNODES * HIDD;
    float* alpha = w; w += (size_t)NEDGES * NHEAD;
    float* amax  = w; w += (size_t)NNODES * NHEAD;
    float* denom = w; w += (size_t)NNODES * NHEAD;
    float* pool  = w; w += (size_t)NB * 2 * HIDD;

    // layer 1: x -> h
    run_layer(x, Wq1,bq1, Wk1,bk1, Wv1,bv1, Ws1,bs1, src, dst,
              q,k,v,s, agg, alpha, amax, denom, h, stream);
    // layer 2: h -> h (in-place safe: h only read by gemm4, rewritten by finalize)
    run_layer(h, Wq2,bq2, Wk2,bk2, Wv2,bv2, Ws2,bs2, src, dst,
              q,k,v,s, agg, alpha, amax, denom, h, stream);

    const int NT = 256;
    pool_init<<<(NB*2*HIDD + NT-1)/NT, NT, 0, stream>>>(pool);
    pool_accumulate<<<(NNODES*HIDD + NT-1)/NT, NT, 0, stream>>>(h, batch, pool);
    classifier<<<1, 128, 0, stream>>>(pool, Wlin, blin, (float*)d_out);
}